// LocalLinear_4398046511709
// MI455X (gfx1250) — compile-verified
//
#include <hip/hip_runtime.h>
#include <hip/hip_bf16.h>

// LocalLinear: out[b,f,l] = sum_k x[b, f+k] * w[f,k,l] + bias[f,l]
// B=256, IN=4096, K=16, FOLD=4081, L=64. Memory-bound (output = 267 MB f32),
// so keep full f32 precision and use V_WMMA_F32_16X16X4_F32 chained x4 (K=16).

#define IN_FEATURES 4096
#define KERNEL_K    16
#define FOLD_N      4081
#define LOCAL_L     64
#define BATCH_N     256

typedef __attribute__((ext_vector_type(2))) float v2f;
typedef __attribute__((ext_vector_type(8))) float v8f;

__global__ __launch_bounds__(128) void local_linear_wmma_f32(
    const float* __restrict__ x,      // [BATCH, IN_FEATURES]
    const float* __restrict__ w,      // [FOLD, KERNEL_K, LOCAL_L]
    const float* __restrict__ bias,   // [FOLD, LOCAL_L]
    float* __restrict__ out)          // [BATCH, FOLD, LOCAL_L]
{
    const int f     = blockIdx.x;          // fold index 0..4080
    const int mbase = blockIdx.y << 4;     // batch tile base (16 rows)
    const int wave  = threadIdx.x >> 5;    // 0..3 -> L tile of 16
    const int lane  = threadIdx.x & 31;
    const int half  = lane >> 4;           // 0: lanes 0-15, 1: lanes 16-31
    const int lid   = lane & 15;
    const int nbase = wave << 4;

    // A-matrix source: row m = mbase+lid of the sliding window, columns f+k.
    const float* xrow = x + (size_t)(mbase + lid) * IN_FEATURES + f;
    // B-matrix source: weight block for this fold.
    const float* wf = w + (size_t)f * (KERNEL_K * LOCAL_L);

    // Prefetch next fold's weight block (streams through L2 fold-to-fold).
    if (f + 1 < FOLD_N)
        __builtin_prefetch(wf + KERNEL_K * LOCAL_L, 0, 1);

    v8f c = {};  // 16x16 f32 accumulator tile

    // K=16 as 4 chained 16x16x4 f32 WMMAs (exact f32 math).
    #pragma unroll
    for (int kc = 0; kc < 4; ++kc) {
        const int k0 = (kc << 2) + (half << 1);   // K for vector element 0
        v2f a, b;
        // 32-bit A 16x4 layout: VGPR j holds K = (lane>>4)*2 + j, row M = lane&15
        a.x = xrow[k0];
        a.y = xrow[k0 + 1];
        // 32-bit B 4x16 layout: VGPR j holds K = (lane>>4)*2 + j, col N = lane&15
        b.x = wf[(size_t)(k0    ) * LOCAL_L + nbase + lid];
        b.y = wf[(size_t)(k0 + 1) * LOCAL_L + nbase + lid];
        // 8 args: (neg_a, A, neg_b, B, c_mod, C, reuse_a, reuse_b)
        c = __builtin_amdgcn_wmma_f32_16x16x4_f32(
                false, a, false, b, (short)0, c, false, false);
    }

    // Bias: same l for all 8 accumulator rows of this lane.
    const float bv = bias[(size_t)f * LOCAL_L + nbase + lid];

    // C/D layout: VGPR r -> row M = r + 8*half, col N = lane&15.
    float* obase = out + ((size_t)(mbase + (half << 3)) * FOLD_N + f) * LOCAL_L
                       + nbase + lid;
    const size_t row_stride = (size_t)FOLD_N * LOCAL_L;  // +1 batch row
    #pragma unroll
    for (int r = 0; r < 8; ++r) {
        obase[(size_t)r * row_stride] = c[r] + bv;
    }
}

extern "C" void kernel_launch(void* const* d_in, const int* in_sizes, int n_in,
                              void* d_out, int out_size, void* d_ws, size_t ws_size,
                              hipStream_t stream) {
    const float* x    = (const float*)d_in[0];
    const float* w    = (const float*)d_in[1];
    const float* bias = (const float*)d_in[2];
    float* out = (float*)d_out;

    dim3 grid(FOLD_N, BATCH_N / 16);  // (fold, batch tile)
    dim3 block(128);                  // 4 waves: one 16-wide L tile each
    local_linear_wmma_f32<<<grid, block, 0, stream>>>(x, w, bias, out);
}